// Unpool_LS_36661840838924
// MI455X (gfx1250) — compile-verified
//
#include <hip/hip_runtime.h>

// Unpool-LS: B=8, H=W=256, C=64, pool 2x2 -> h=w=128, L=4.
// Memory-bound streaming kernel: one thread handles one (b,h,w) block for 4
// channels (float4), so all global traffic is coalesced b128 with NT hints.

typedef float v4f __attribute__((ext_vector_type(4)));

#define B_   8
#define H_   256
#define W_   256
#define C_   64
#define h_   128
#define w_   128
#define C4_  16                      // float4 per pixel
#define TOT_V4 (B_ * h_ * w_ * C4_) // 2,097,152 vec4 tasks

// Solve one 4-element block + pooled value p.
// a0..a3 are the block values in row-major-within-block order (l = 0..3).
// Outputs: o* replaced values, q* mask values, rp replacement average.
__device__ __forceinline__ void solve1(float a0, float a1, float a2, float a3, float p,
                                       float& o0, float& o1, float& o2, float& o3,
                                       float& q0, float& q1, float& q2, float& q3,
                                       float& rp)
{
    // Stable descending rank: rank[i] = #{j: a[j] > a[i]} + #{j<i: a[j] == a[i]}
    int r0 = (int)(a1 > a0)  + (int)(a2 > a0)  + (int)(a3 > a0);
    int r1 = (int)(a0 >= a1) + (int)(a2 > a1)  + (int)(a3 > a1);
    int r2 = (int)(a0 >= a2) + (int)(a1 >= a2) + (int)(a3 > a2);
    int r3 = (int)(a0 >= a3) + (int)(a1 >= a3) + (int)(a2 >= a3);

    // Rank-select the descending-sorted values (branch-free cndmask chains).
    float sv0 = (r0 == 0) ? a0 : (r1 == 0) ? a1 : (r2 == 0) ? a2 : a3;
    float sv1 = (r0 == 1) ? a0 : (r1 == 1) ? a1 : (r2 == 1) ? a2 : a3;
    float sv2 = (r0 == 2) ? a0 : (r1 == 2) ? a1 : (r2 == 2) ? a2 : a3;
    float sv3 = (r0 == 3) ? a0 : (r1 == 3) ? a1 : (r2 == 3) ? a2 : a3;

    // Cumulative sums seeded with pooled value, candidate averages.
    float c0 = p  + sv0;
    float c1 = c0 + sv1;
    float c2 = c1 + sv2;
    float c3 = c2 + sv3;
    float g0 = c0 / 2.0f;   // exact (power of two)
    float g1 = c1 / 3.0f;   // IEEE divide, matches jnp
    float g2 = c2 / 4.0f;   // exact
    float g3 = c3 / 5.0f;   // IEEE divide

    // First-max argmax (strict > keeps earliest index on ties, like jnp.argmax).
    float best = g0; int k = 0;
    if (g1 > best) { best = g1; k = 1; }
    if (g2 > best) { best = g2; k = 2; }
    if (g3 > best) { best = g3; k = 3; }

    // (k+1)/(k+2) as correctly-rounded float constants.
    float frac = (k == 0) ? 0.5f : (k == 1) ? (2.0f / 3.0f) : (k == 2) ? 0.75f : 0.8f;

    o0 = (r0 <= k) ? best : a0;  q0 = (r0 <= k) ? frac : 1.0f;
    o1 = (r1 <= k) ? best : a1;  q1 = (r1 <= k) ? frac : 1.0f;
    o2 = (r2 <= k) ? best : a2;  q2 = (r2 <= k) ? frac : 1.0f;
    o3 = (r3 <= k) ? best : a3;  q3 = (r3 <= k) ? frac : 1.0f;
    rp = best;
}

__global__ void __launch_bounds__(256)
Unpool_LS_36661840838924_kernel(const v4f* __restrict__ x0,
                                const v4f* __restrict__ x1,
                                v4f* __restrict__ out0,   // [B,H,W,C] replaced
                                v4f* __restrict__ out1,   // [B,h,w,C] repl value
                                v4f* __restrict__ out2)   // [B,H,W,C] mask
{
    unsigned t = blockIdx.x * blockDim.x + threadIdx.x;
    if (t >= TOT_V4) return;

    // t = ((b*h + hq)*w + wq)*C4 + c4  -> also the flat index into x1/out1.
    unsigned c4 = t & 15u;
    unsigned wq = (t >> 4) & 127u;
    unsigned hq = (t >> 11) & 127u;
    unsigned b  = t >> 18;

    unsigned y = hq * 2u;
    unsigned x = wq * 2u;

    const size_t rowStride = (size_t)W_ * C4_;                       // vec4 per full-res row
    const size_t p00 = ((size_t)(b * H_ + y) * W_ + x) * C4_ + c4;   // (y, x)
    const size_t p01 = p00 + C4_;                                    // (y, x+1)
    const size_t p10 = p00 + rowStride;                              // (y+1, x)
    const size_t p11 = p10 + C4_;                                    // (y+1, x+1)

    // gfx1250 prefetch path: pull the next block-row's lines toward L2 early.
    if (y + 2u < H_) {
        __builtin_prefetch((const void*)&x0[p00 + 2u * rowStride], 0, 0);
        __builtin_prefetch((const void*)&x0[p10 + 2u * rowStride], 0, 0);
    }

    // Streaming reads: each element read exactly once -> non-temporal b128.
    v4f v0 = __builtin_nontemporal_load(&x0[p00]);
    v4f v1 = __builtin_nontemporal_load(&x0[p01]);
    v4f v2 = __builtin_nontemporal_load(&x0[p10]);
    v4f v3 = __builtin_nontemporal_load(&x0[p11]);
    v4f pl = __builtin_nontemporal_load(&x1[t]);

    v4f O0, O1, O2, O3, M0, M1, M2, M3, R;
#pragma unroll
    for (int i = 0; i < 4; ++i) {
        float o0, o1, o2, o3, q0, q1, q2, q3, rp;
        solve1(v0[i], v1[i], v2[i], v3[i], pl[i],
               o0, o1, o2, o3, q0, q1, q2, q3, rp);
        O0[i] = o0; O1[i] = o1; O2[i] = o2; O3[i] = o3;
        M0[i] = q0; M1[i] = q1; M2[i] = q2; M3[i] = q3;
        R[i]  = rp;
    }

    // Streaming writes: written exactly once -> non-temporal b128.
    __builtin_nontemporal_store(O0, &out0[p00]);
    __builtin_nontemporal_store(O1, &out0[p01]);
    __builtin_nontemporal_store(O2, &out0[p10]);
    __builtin_nontemporal_store(O3, &out0[p11]);

    __builtin_nontemporal_store(M0, &out2[p00]);
    __builtin_nontemporal_store(M1, &out2[p01]);
    __builtin_nontemporal_store(M2, &out2[p10]);
    __builtin_nontemporal_store(M3, &out2[p11]);

    __builtin_nontemporal_store(R, &out1[t]);
}

extern "C" void kernel_launch(void* const* d_in, const int* in_sizes, int n_in,
                              void* d_out, int out_size, void* d_ws, size_t ws_size,
                              hipStream_t stream) {
    (void)in_sizes; (void)n_in; (void)out_size; (void)d_ws; (void)ws_size;

    const v4f* x0 = (const v4f*)d_in[0];   // [8,256,256,64] f32
    const v4f* x1 = (const v4f*)d_in[1];   // [8,128,128,64] f32

    float* out = (float*)d_out;
    const size_t n_full   = (size_t)B_ * H_ * W_ * C_;   // 33,554,432
    const size_t n_pooled = (size_t)B_ * h_ * w_ * C_;   //  8,388,608
    v4f* out0 = (v4f*)out;                               // replaced map
    v4f* out1 = (v4f*)(out + n_full);                    // repl values
    v4f* out2 = (v4f*)(out + n_full + n_pooled);         // mask map

    dim3 block(256);                       // 8 wave32 per workgroup
    dim3 grid((TOT_V4 + 255) / 256);       // 8192 blocks
    Unpool_LS_36661840838924_kernel<<<grid, block, 0, stream>>>(x0, x1, out0, out1, out2);
}